// SpikeFP32ScaleBy2K_76312978916075
// MI455X (gfx1250) — compile-verified
//
#include <hip/hip_runtime.h>
#include <cstdint>

// SpikeFP32ScaleBy2K: rows of 32 "bit floats" (0.0/1.0) encoding an fp32 word
// MSB-first. out = x with exponent byte replaced by (e_x + trunc-int(k)) mod 256,
// unless k == +/-0 in which case out = x.
//
// Streaming kernel: 256B in + 128B out per row, 768MB total -> ~33us at 23.3TB/s.
// Data path: CDNA5 Tensor Data Mover. One tensor_load_to_lds per 32KB tile
// (issued by wave 0, TENSORcnt-tracked, double buffered). The TDM pad feature
// (pad every 32 dwords by 4 dwords) lays rows out with a 144B LDS stride so the
// per-thread ds_load_b128 row reads are a balanced 2-way bank pattern.
// Fallback path: GLOBAL_LOAD_ASYNC_TO_LDS_B128 inline asm.

#if defined(__has_builtin)
#  if __has_builtin(__builtin_amdgcn_tensor_load_to_lds)
#    define USE_TDM 1
#  endif
#endif
#ifndef USE_TDM
#  define USE_TDM 0
#endif

namespace {

typedef uint32_t u32x4 __attribute__((ext_vector_type(4)));
typedef int      i32x8 __attribute__((ext_vector_type(8)));
typedef int      i32x4 __attribute__((ext_vector_type(4)));

constexpr int kTileRows  = 256;
constexpr int kThreads   = 256;
constexpr int kRowFloats = 32;
constexpr int kTileElems = kTileRows * kRowFloats;  // 8192 dwords = 32KB
constexpr int kLdsStride = 144;                     // 128B row + 16B TDM pad
constexpr int kBufBytes  = kTileRows * kLdsStride;  // 36864 B per buffer

__device__ __forceinline__ uint32_t bit01(uint32_t u) {
  // u is 0x00000000 (0.0f) or 0x3F800000 (1.0f) -> 0 or 1
  return u >> 29;
}

#if USE_TDM
// Issue one TDM load: 32KB contiguous tile -> padded LDS layout.
// D# group0: count=1 | lds_addr | global_addr(57b) | type=2
// D# group1: data_size=4B, pad_enable, pad_interval=32 dwords (code 4),
//            pad_amount=4 dwords (code 3); 1-D tensor: dim0=tile_dim0=8192.
__device__ __forceinline__ void tdm_issue_tile(const float* gtile, uint32_t lds_base) {
  const uint64_t ga = (uint64_t)(uintptr_t)gtile;
  u32x4 g0;
  g0[0] = 1u;                                    // count=1, user descriptor
  g0[1] = lds_base;                              // lds_addr (bytes)
  g0[2] = (uint32_t)ga;                          // global_addr[31:0]
  g0[3] = (uint32_t)(ga >> 32) | (2u << 30);     // global_addr[56:32], type=2
  i32x8 g1;
  g1[0] = (int)((2u << 16) |                     // data_size = 4B
                (1u << 20) |                     // pad_enable
                (4u << 22) |                     // pad_interval: 32 dwords
                (3u << 25));                     // pad_amount: 4 dwords (16B)
  g1[1] = (int)(((uint32_t)kTileElems & 0xFFFFu) << 16);    // tensor_dim0[15:0]
  g1[2] = (int)(((uint32_t)kTileElems >> 16) | (1u << 16)); // dim0[31:16], tensor_dim1=1
  g1[3] = (int)(((uint32_t)kTileElems & 0xFFFFu) << 16);    // tile_dim0 = 8192
  g1[4] = 1;                                     // tile_dim1 = 1, tile_dim2 = 0
  g1[5] = (int)kTileElems;                       // tensor_dim0_stride[31:0]
  g1[6] = 0;
  g1[7] = 0;
  const i32x4 z4 = {0, 0, 0, 0};                 // groups 2/3 unused (<=2D tensor)
  const i32x8 z8 = {0, 0, 0, 0, 0, 0, 0, 0};
  __builtin_amdgcn_tensor_load_to_lds(g0, g1, z4, z4, z8, 0);
}

__device__ __forceinline__ void wait_tensorcnt_le2() {
#if defined(__has_builtin) && __has_builtin(__builtin_amdgcn_s_wait_tensorcnt)
  __builtin_amdgcn_s_wait_tensorcnt(2);
#else
  asm volatile("s_wait_tensorcnt 2" ::: "memory");
#endif
}
__device__ __forceinline__ void wait_tensorcnt_le0() {
#if defined(__has_builtin) && __has_builtin(__builtin_amdgcn_s_wait_tensorcnt)
  __builtin_amdgcn_s_wait_tensorcnt(0);
#else
  asm volatile("s_wait_tensorcnt 0" ::: "memory");
#endif
}
#else
// Fallback: per-lane async copies, lane-contiguous 512B per wave-instruction,
// scattered into the padded LDS layout.
__device__ __forceinline__ void async_tile_copy(const float* gtile,
                                                uint32_t lds_base, int tid) {
#pragma unroll
  for (int j = 0; j < 8; ++j) {
    const int c   = j * kThreads + tid;  // 16B-chunk id within tile, 0..2047
    const int row = c >> 3;
    const int col = c & 7;
    const uint32_t voff  = (uint32_t)c * 16u;
    const uint32_t laddr = lds_base + (uint32_t)(row * kLdsStride + col * 16);
    asm volatile("global_load_async_to_lds_b128 %0, %1, %2"
                 :: "v"(laddr), "v"(voff), "s"(gtile)
                 : "memory");
  }
}
#endif

// Core bit math for one row. xv/kv hold the 32 raw words each.
__device__ __forceinline__ void scale_row(const uint4 (&xv)[8],
                                          const uint4 (&kv)[8],
                                          uint4 (&o)[8]) {
  const uint32_t ex =
      (bit01(xv[0].y) << 7) | (bit01(xv[0].z) << 6) | (bit01(xv[0].w) << 5) |
      (bit01(xv[1].x) << 4) | (bit01(xv[1].y) << 3) | (bit01(xv[1].z) << 2) |
      (bit01(xv[1].w) << 1) |  bit01(xv[2].x);
  const uint32_t sk = bit01(kv[0].x);
  const uint32_t ek =
      (bit01(kv[0].y) << 7) | (bit01(kv[0].z) << 6) | (bit01(kv[0].w) << 5) |
      (bit01(kv[1].x) << 4) | (bit01(kv[1].y) << 3) | (bit01(kv[1].z) << 2) |
      (bit01(kv[1].w) << 1) |  bit01(kv[2].x);
  const uint32_t mk7 =
      (bit01(kv[2].y) << 6) | (bit01(kv[2].z) << 5) | (bit01(kv[2].w) << 4) |
      (bit01(kv[3].x) << 3) | (bit01(kv[3].y) << 2) | (bit01(kv[3].z) << 1) |
       bit01(kv[3].w);

  // k == +/-0  <=>  elements 1..31 of k are all zero (sign excluded)
  uint32_t nz = kv[0].y | kv[0].z | kv[0].w;
#pragma unroll
  for (int j = 1; j < 8; ++j) nz |= kv[j].x | kv[j].y | kv[j].z | kv[j].w;
  const bool kzero = (nz == 0u);

  // Matches the reference's 8-bit two's-complement circuit exactly (mod-256
  // wraps, shift amount uses only low 3 bits of (e_k - 127)).
  const uint32_t E8    = (ek + 129u) & 0xFFu;       // (e_k - 127) mod 256
  const uint32_t shamt = 7u - (E8 & 7u);
  const uint32_t kabs  = (0x80u | mk7) >> shamt;    // trunc(|k|) (wrapped)
  const uint32_t kneg  = (0u - kabs) & 0xFFu;
  const uint32_t kfin  = sk ? kneg : kabs;
  const uint32_t enew  = (ex + kfin) & 0xFFu;

  uint32_t eb[8];
#pragma unroll
  for (int j = 0; j < 8; ++j)
    eb[j] = ((enew >> (7 - j)) & 1u) * 0x3F800000u; // bit -> 1.0f/0.0f pattern

  o[0].x = xv[0].x;                    // sign passthrough
  o[0].y = kzero ? xv[0].y : eb[0];
  o[0].z = kzero ? xv[0].z : eb[1];
  o[0].w = kzero ? xv[0].w : eb[2];
  o[1].x = kzero ? xv[1].x : eb[3];
  o[1].y = kzero ? xv[1].y : eb[4];
  o[1].z = kzero ? xv[1].z : eb[5];
  o[1].w = kzero ? xv[1].w : eb[6];
  o[2]   = xv[2];
  o[2].x = kzero ? xv[2].x : eb[7];
#pragma unroll
  for (int j = 3; j < 8; ++j) o[j] = xv[j];  // mantissa passthrough
}

__global__ __launch_bounds__(kThreads) void scale2k_main(
    const float* __restrict__ x, const float* __restrict__ k,
    float* __restrict__ out, int tiles) {
  __shared__ __align__(16) uint8_t smem[4 * kBufBytes];  // x0,k0,x1,k1
  const uint32_t lds0 = (uint32_t)(uintptr_t)(void*)smem;
  const int tid    = threadIdx.x;
  const int stride = gridDim.x;

  int tile = blockIdx.x;
  if (tile >= tiles) return;

#if USE_TDM
  const bool issuer = (tid < 32);  // wave 0 drives the TDM
  if (issuer) {
    tdm_issue_tile(x + (size_t)tile * kTileElems, lds0 + 0u * kBufBytes);
    tdm_issue_tile(k + (size_t)tile * kTileElems, lds0 + 1u * kBufBytes);
  }
#else
  async_tile_copy(x + (size_t)tile * kTileElems, lds0 + 0u * kBufBytes, tid);
  async_tile_copy(k + (size_t)tile * kTileElems, lds0 + 1u * kBufBytes, tid);
#endif

  int buf = 0;
  for (; tile < tiles; tile += stride) {
    const int nxt = tile + stride;
#if USE_TDM
    if (issuer) {
      if (nxt < tiles) {
        const uint32_t nb = (uint32_t)(buf ^ 1);
        tdm_issue_tile(x + (size_t)nxt * kTileElems, lds0 + (2u * nb + 0u) * kBufBytes);
        tdm_issue_tile(k + (size_t)nxt * kTileElems, lds0 + (2u * nb + 1u) * kBufBytes);
        // 2 newest TDM ops outstanding (next tile); TDM ops from one wave
        // complete in order, so <=2 means the current tile has landed.
        wait_tensorcnt_le2();
      } else {
        wait_tensorcnt_le0();
      }
    }
#else
    if (nxt < tiles) {
      const uint32_t nb = (uint32_t)(buf ^ 1);
      async_tile_copy(x + (size_t)nxt * kTileElems, lds0 + (2u * nb + 0u) * kBufBytes, tid);
      async_tile_copy(k + (size_t)nxt * kTileElems, lds0 + (2u * nb + 1u) * kBufBytes, tid);
      asm volatile("s_wait_asynccnt 16" ::: "memory");
    } else {
      asm volatile("s_wait_asynccnt 0" ::: "memory");
    }
#endif
    __syncthreads();  // current tile visible to all waves

    const uint8_t* xrow = smem + (2 * buf + 0) * kBufBytes + tid * kLdsStride;
    const uint8_t* krow = smem + (2 * buf + 1) * kBufBytes + tid * kLdsStride;
    uint4 xv[8], kv[8], o[8];
#pragma unroll
    for (int j = 0; j < 8; ++j) xv[j] = *(const uint4*)(xrow + 16 * j);
#pragma unroll
    for (int j = 0; j < 8; ++j) kv[j] = *(const uint4*)(krow + 16 * j);

    scale_row(xv, kv, o);

    uint4* orow = (uint4*)(out + ((size_t)tile * kTileRows + tid) * kRowFloats);
#pragma unroll
    for (int j = 0; j < 8; ++j) orow[j] = o[j];

    __syncthreads();  // everyone done reading this buffer before it is re-filled
    buf ^= 1;
  }
}

// Fallback for any rows not covered by full tiles (N=2M is exact, but be safe).
__global__ void scale2k_tail(const float* __restrict__ x, const float* __restrict__ k,
                             float* __restrict__ out, size_t row0, size_t rows) {
  const size_t r = row0 + (size_t)blockIdx.x * blockDim.x + threadIdx.x;
  if (r >= rows) return;
  uint4 xv[8], kv[8], o[8];
  const uint4* xp = (const uint4*)(x + r * kRowFloats);
  const uint4* kp = (const uint4*)(k + r * kRowFloats);
#pragma unroll
  for (int j = 0; j < 8; ++j) { xv[j] = xp[j]; kv[j] = kp[j]; }
  scale_row(xv, kv, o);
  uint4* op = (uint4*)(out + r * kRowFloats);
#pragma unroll
  for (int j = 0; j < 8; ++j) op[j] = o[j];
}

}  // namespace

extern "C" void kernel_launch(void* const* d_in, const int* in_sizes, int n_in,
                              void* d_out, int out_size, void* d_ws, size_t ws_size,
                              hipStream_t stream) {
  const float* x = (const float*)d_in[0];
  const float* k = (const float*)d_in[1];
  float* out = (float*)d_out;

  const size_t rows  = (size_t)in_sizes[0] / kRowFloats;  // 2,097,152
  const int    tiles = (int)(rows / kTileRows);           // 8192

  if (tiles > 0) {
    const int blocks = tiles < 2048 ? tiles : 2048;       // ~4 tiles per block, pipelined
    scale2k_main<<<blocks, kThreads, 0, stream>>>(x, k, out, tiles);
  }
  const size_t done = (size_t)tiles * kTileRows;
  if (done < rows) {
    const size_t rem = rows - done;
    const int b = (int)((rem + 255) / 256);
    scale2k_tail<<<b, 256, 0, stream>>>(x, k, out, done, rows);
  }
}